// npllLoss_3350074491170
// MI455X (gfx1250) — compile-verified
//
#include <hip/hip_runtime.h>
#include <hip/hip_bf16.h>
#include <stdint.h>

// ---------------------------------------------------------------------------
// Cox partial likelihood (Efron ties), bucketed by integer time.
//   NT   = number of time buckets (T_MAX in the reference)
//   NTP  = padded bucket count (25 chunks * 4096) so every kernel sees full,
//          zero-padded chunks (zeros do not perturb suffix sums).
// ---------------------------------------------------------------------------
#define NT      100000
#define CHUNK   4096
#define NCHUNK  25
#define NTP     (NCHUNK * CHUNK)   // 102400

typedef float v2f __attribute__((ext_vector_type(2)));
typedef float v8f __attribute__((ext_vector_type(8)));

// ---------------------------------------------------------------------------
// K0: zero the workspace accumulators (d_ws is poisoned, not re-zeroed).
// ---------------------------------------------------------------------------
__global__ void zero_ws_kernel(float* __restrict__ ws, int nwords) {
  int i = blockIdx.x * blockDim.x + threadIdx.x;
  if (i < nwords) ws[i] = 0.0f;
}

// ---------------------------------------------------------------------------
// K1: streaming histogram pass.  For each sample:
//   grp_hz[t] += exp(lh);  if event: m[t]+=1, lognom[t]+=lh, ties[t]+=exp(lh)
// ---------------------------------------------------------------------------
__global__ void pass1_hist(const float* __restrict__ lh,
                           const int*   __restrict__ tg,
                           const int*   __restrict__ ev,
                           float* __restrict__ grp_hz,
                           float* __restrict__ mcnt,
                           float* __restrict__ lognom,
                           float* __restrict__ ties,
                           int n4, int n) {
  int i = blockIdx.x * blockDim.x + threadIdx.x;
  if (i < n4) {
    float4 l4 = ((const float4*)lh)[i];
    int4   t4 = ((const int4*)tg)[i];
    int4   e4 = ((const int4*)ev)[i];
    float ls[4] = {l4.x, l4.y, l4.z, l4.w};
    int   ts[4] = {t4.x, t4.y, t4.z, t4.w};
    int   es[4] = {e4.x, e4.y, e4.z, e4.w};
#pragma unroll
    for (int k = 0; k < 4; ++k) {
      float h = expf(ls[k]);
      int   t = ts[k];
      atomicAdd(&grp_hz[t], h);
      if (es[k] != 0) {
        atomicAdd(&mcnt[t],   1.0f);
        atomicAdd(&lognom[t], ls[k]);
        atomicAdd(&ties[t],   h);
      }
    }
  }
  // scalar tail (n not multiple of 4) -- handled by one thread, normally empty
  if (i == 0) {
    for (int k = n4 * 4; k < n; ++k) {
      float l = lh[k];
      float h = expf(l);
      int   t = tg[k];
      atomicAdd(&grp_hz[t], h);
      if (ev[k] != 0) {
        atomicAdd(&mcnt[t],   1.0f);
        atomicAdd(&lognom[t], l);
        atomicAdd(&ties[t],   h);
      }
    }
  }
}

// ---------------------------------------------------------------------------
// K2: per-chunk totals (25 blocks x 4096 elements).
// ---------------------------------------------------------------------------
__global__ void chunk_reduce(const float* __restrict__ grp_hz,
                             float* __restrict__ chunkSum) {
  __shared__ float red[256];
  const int tid = threadIdx.x;
  const float4* p = (const float4*)(grp_hz + (size_t)blockIdx.x * CHUNK);
  float s = 0.0f;
#pragma unroll
  for (int k = 0; k < 4; ++k) {          // 256 threads * 4 float4 = 4096
    float4 v = p[tid + 256 * k];
    s += v.x + v.y + v.z + v.w;
  }
  red[tid] = s;
  __syncthreads();
  for (int off = 128; off > 0; off >>= 1) {
    if (tid < off) red[tid] += red[tid + off];
    __syncthreads();
  }
  if (tid == 0) chunkSum[blockIdx.x] = red[0];
}

// ---------------------------------------------------------------------------
// K3: exclusive suffix scan over the 25 chunk sums (tiny, serial).
// ---------------------------------------------------------------------------
__global__ void chunk_scan(const float* __restrict__ chunkSum,
                           float* __restrict__ chunkOff) {
  if (threadIdx.x == 0 && blockIdx.x == 0) {
    float s = 0.0f;
    for (int b = NCHUNK - 1; b >= 0; --b) { chunkOff[b] = s; s += chunkSum[b]; }
  }
}

// ---------------------------------------------------------------------------
// K4: in-chunk suffix scan via WMMA (matmul-as-scan) + async LDS staging.
//
// Each wave scans a 256-element sub-chunk laid out column-major as X (16x16):
//   X[k][n] = x[n*16 + k].
// S = U * X with U upper-triangular-ones gives S[m][n] = sum_{k>=m} X[k][n],
// i.e. 16 independent 16-element suffix scans.  Built from 4 chained
// V_WMMA_F32_16X16X4_F32 ops (K-slices j = 0..3).
//
// Operand layouts per CDNA5 ISA 7.12.2 (wave32):
//   A 16x4 : lane 0-15 -> M=lane,    VGPR0=K0, VGPR1=K1
//            lane16-31 -> M=lane-16, VGPR0=K2, VGPR1=K3
//   B 4x16 : VGPR0 = rows {0 (lanes 0-15), 2 (lanes 16-31)}
//            VGPR1 = rows {1, 3}                           (row striped/lane)
//   C/D    : VGPR r = rows {r (lanes 0-15), r+8 (lanes 16-31)}, col = lane&15
// ---------------------------------------------------------------------------
__global__ void __launch_bounds__(512)
suffix_scan_wmma(const float* __restrict__ grp_hz,
                 const float* __restrict__ chunkOff,
                 float* __restrict__ denom) {
  __shared__ float xbuf[CHUNK];       // 16 KB chunk staging
  __shared__ float colsum[16 * 16];   // per-wave 16 column sums
  __shared__ float subTot[16];
  __shared__ float subOff[16];

  const int tid   = threadIdx.x;
  const int lane  = tid & 31;
  const int wave  = tid >> 5;        // 16 waves of 32
  const int cbase = blockIdx.x * CHUNK;

  // ---- CDNA5 async global->LDS staging (ASYNCcnt) -------------------------
  {
    unsigned long long g0 = (unsigned long long)(uintptr_t)(grp_hz + cbase + tid * 4);
    unsigned long long g1 = (unsigned long long)(uintptr_t)(grp_hz + cbase + 2048 + tid * 4);
    // generic->LDS offset: low 32 bits of the flat address (ISA 10.2)
    uint32_t l0 = (uint32_t)(uintptr_t)&xbuf[tid * 4];
    uint32_t l1 = (uint32_t)(uintptr_t)&xbuf[2048 + tid * 4];
    asm volatile("global_load_async_to_lds_b128 %0, %1, off"
                 :: "v"(l0), "v"(g0) : "memory");
    asm volatile("global_load_async_to_lds_b128 %0, %1, off"
                 :: "v"(l1), "v"(g1) : "memory");
    asm volatile("s_wait_asynccnt 0x0" ::: "memory");
  }
  __syncthreads();

  const int hh = lane >> 4;          // half: 0 -> lanes 0-15, 1 -> lanes 16-31
  const int n  = lane & 15;          // A: row M;  B/D: column N
  const float* X = &xbuf[wave * 256];

  // ---- S = U * X via 4 chained f32 WMMAs (EXEC all-ones: no divergence) ---
  v8f c = {};                        // f32 accumulator, 8 VGPRs
#pragma unroll
  for (int j = 0; j < 4; ++j) {
    const int kk = 4 * j + 2 * hh;
    v2f a, b;
    a.x = (kk     >= n) ? 1.0f : 0.0f;   // U[m][kk],   m == lane&15
    a.y = (kk + 1 >= n) ? 1.0f : 0.0f;   // U[m][kk+1]
    b.x = X[n * 16 + kk];                // X[kk  ][n]
    b.y = X[n * 16 + kk + 1];            // X[kk+1][n]
    c = __builtin_amdgcn_wmma_f32_16x16x4_f32(
        /*neg_a=*/false, a, /*neg_b=*/false, b,
        /*c_mod=*/(short)0, c, /*reuse_a=*/false, /*reuse_b=*/false);
  }

  // ---- column totals (row M=0 = c[0] on lanes 0-15) -----------------------
  if (hh == 0) colsum[wave * 16 + n] = c[0];
  __syncthreads();

  float coff = 0.0f;                               // cols strictly right of n
  for (int q = n + 1; q < 16; ++q) coff += colsum[wave * 16 + q];
  if (lane == 0) subTot[wave] = coff + colsum[wave * 16];   // sub-chunk total
  __syncthreads();

  if (tid == 0) {                                  // suffix over 16 sub-chunks
    float s = 0.0f;
    for (int w = 15; w >= 0; --w) { subOff[w] = s; s += subTot[w]; }
  }
  __syncthreads();

  const float addv = coff + subOff[wave] + chunkOff[blockIdx.x];

  // element index = n*16 + (r + 8*hh): 8 consecutive floats -> two B128 stores
  float4 o0 = make_float4(c[0] + addv, c[1] + addv, c[2] + addv, c[3] + addv);
  float4 o1 = make_float4(c[4] + addv, c[5] + addv, c[6] + addv, c[7] + addv);
  float4* out = (float4*)(denom + cbase + wave * 256 + n * 16 + 8 * hh);
  out[0] = o0;
  out[1] = o1;
}

// ---------------------------------------------------------------------------
// K5: Efron per-bucket terms, block-reduced then globally accumulated.
//   pll_t = lognom_t - sum_{l=0}^{m-1} log(denom_t - (l/m)*ties_t)
// ---------------------------------------------------------------------------
__global__ void efron_terms(const float* __restrict__ mcnt,
                            const float* __restrict__ lognom,
                            const float* __restrict__ ties,
                            const float* __restrict__ denom,
                            float* __restrict__ accum) {
  __shared__ float acc[2];
  if (threadIdx.x == 0) { acc[0] = 0.0f; acc[1] = 0.0f; }
  __syncthreads();

  int t = blockIdx.x * blockDim.x + threadIdx.x;
  if (t < NT) {
    float mj = mcnt[t];
    if (mj > 0.0f) {
      float D   = denom[t];
      float T   = ties[t];
      float inv = 1.0f / mj;
      float ld  = 0.0f;
      int   mi  = (int)mj;
      for (int l = 0; l < mi; ++l)
        ld += logf(D - ((float)l * inv) * T);
      atomicAdd(&acc[0], lognom[t] - ld);
      atomicAdd(&acc[1], 1.0f);
    }
  }
  __syncthreads();
  if (threadIdx.x == 0) {
    atomicAdd(&accum[0], acc[0]);
    atomicAdd(&accum[1], acc[1]);
  }
}

// ---------------------------------------------------------------------------
// K6: loss = -(sum pll) / (num buckets with events)
// ---------------------------------------------------------------------------
__global__ void finalize_kernel(const float* __restrict__ accum,
                                float* __restrict__ out) {
  if (threadIdx.x == 0 && blockIdx.x == 0)
    out[0] = -(accum[0] / accum[1]);
}

// ---------------------------------------------------------------------------
extern "C" void kernel_launch(void* const* d_in, const int* in_sizes, int n_in,
                              void* d_out, int out_size, void* d_ws, size_t ws_size,
                              hipStream_t stream) {
  const float* lh = (const float*)d_in[0];   // log hazards  (float32, N)
  const int*   tg = (const int*)d_in[1];     // times        (int32,  N)
  const int*   ev = (const int*)d_in[2];     // events       (int32,  N)
  const int    n  = in_sizes[0];

  // workspace layout (floats): 5 padded bucket arrays + small scratch ~2.05 MB
  float* ws       = (float*)d_ws;
  float* grp_hz   = ws;               // [NTP] sum exp per bucket
  float* mcnt     = ws + 1 * NTP;     // [NTP] events per bucket (as float)
  float* lognom   = ws + 2 * NTP;     // [NTP] sum lh over events
  float* ties     = ws + 3 * NTP;     // [NTP] sum exp over events
  float* denom    = ws + 4 * NTP;     // [NTP] risk-set suffix sums
  float* chunkSum = ws + 5 * NTP;     // [32]
  float* chunkOff = chunkSum + 32;    // [32]
  float* accum    = chunkOff + 32;    // [2]: pll sum, bucket count

  const int zwords = 5 * NTP + 66;
  zero_ws_kernel<<<(zwords + 255) / 256, 256, 0, stream>>>(ws, zwords);

  const int n4 = n >> 2;
  pass1_hist<<<(n4 + 255) / 256, 256, 0, stream>>>(lh, tg, ev,
                                                   grp_hz, mcnt, lognom, ties,
                                                   n4, n);

  chunk_reduce<<<NCHUNK, 256, 0, stream>>>(grp_hz, chunkSum);
  chunk_scan<<<1, 32, 0, stream>>>(chunkSum, chunkOff);
  suffix_scan_wmma<<<NCHUNK, 512, 0, stream>>>(grp_hz, chunkOff, denom);

  efron_terms<<<(NT + 255) / 256, 256, 0, stream>>>(mcnt, lognom, ties, denom, accum);
  finalize_kernel<<<1, 32, 0, stream>>>(accum, (float*)d_out);
}